// MolFusion_20761871909702
// MI455X (gfx1250) — compile-verified
//
#include <hip/hip_runtime.h>
#include <hip/hip_bf16.h>

typedef __attribute__((ext_vector_type(2))) float v2f;
typedef __attribute__((ext_vector_type(8))) float v8f;

__device__ __forceinline__ void atomicAddF(float* p, float v) {
    unsafeAtomicAdd(p, v);  // hardware global_atomic_add_f32
}

// ---------------------------------------------------------------------------
// C[M,128] = A[M,K] @ W[K,128], fp32 WMMA 16x16x4. K % 32 == 0, M % 16 == 0.
// blockDim = 256 (8 waves); each wave computes one 16-row tile x full N=128.
// LDS holds a 32x128 slab of W, k-pair interleaved so each B fragment is one
// aligned ds_load_b64: Wlds[(k>>1)*256 + n*2 + (k&1)].
// ---------------------------------------------------------------------------
__global__ void __launch_bounds__(256)
mf_gemm_wmma(const float* __restrict__ A, const float* __restrict__ W,
             float* __restrict__ C, int M, int K) {
    __shared__ float Wlds[32 * 128];  // 16 KB K-slab of W (interleaved layout)
    const int t = threadIdx.x;
    const int wave = t >> 5;
    const int lane = t & 31;
    const int mbase = (blockIdx.x * 8 + wave) * 16;
    const bool active = (mbase < M);      // wave-uniform at runtime -> EXEC all-1s
    const int mrow = lane & 15;
    const int khalf = lane >> 4;          // lanes 16-31 hold K+2 of the A/B frags

    v8f acc[8] = {};

    for (int k0 = 0; k0 < K; k0 += 32) {
        __syncthreads();
        #pragma unroll
        for (int i = 0; i < 16; ++i) {    // 4096 floats / 256 threads
            int idx = t + i * 256;
            int r = idx >> 7;             // 0..31 (k within slab)
            int n = idx & 127;
            Wlds[(r >> 1) * 256 + n * 2 + (r & 1)] =
                W[(size_t)(k0 + r) * 128 + n];
        }
        __syncthreads();
        if (active) {
            // A 16x4 f32 layout: lane m (0-15) v0/v1 = K0,K1; lanes 16-31 = K2,K3
            const float* Arow = A + (size_t)(mbase + mrow) * K + k0 + 2 * khalf;
            #pragma unroll
            for (int kk = 0; kk < 32; kk += 4) {
                const v2f a = *(const v2f*)(Arow + kk);          // global_load_b64
                const int p = ((kk >> 1) + khalf) * 256 + mrow * 2;
                #pragma unroll
                for (int j = 0; j < 8; ++j) {
                    // B 4x16 layout mirrors A: lane n, v0/v1 = rows (2*khalf, 2*khalf+1)
                    const v2f b = *(const v2f*)(&Wlds[p + j * 32]); // ds_load_b64
                    acc[j] = __builtin_amdgcn_wmma_f32_16x16x4_f32(
                        false, a, false, b, (short)0, acc[j], false, false);
                }
            }
        }
    }
    if (active) {
        // C/D layout: VGPR r, lanes 0-15 -> M=r, N=lane; lanes 16-31 -> M=8+r
        #pragma unroll
        for (int j = 0; j < 8; ++j) {
            #pragma unroll
            for (int r = 0; r < 8; ++r) {
                C[(size_t)(mbase + khalf * 8 + r) * 128 + j * 16 + mrow] = acc[j][r];
            }
        }
    }
}

// ---------------------------------------------------------------------------
// Elementwise / graph kernels
// ---------------------------------------------------------------------------
__global__ void mf_fill(float* p, float v, long n) {
    long i = (long)blockIdx.x * 256 + threadIdx.x;
    if (i < n) p[i] = v;
}

__global__ void mf_edge_deg(const int* __restrict__ dst, float* deg, int E) {
    int e = blockIdx.x * 256 + threadIdx.x;
    if (e < E) atomicAddF(&deg[dst[e]], 1.0f);
}

__global__ void mf_rsqrt(float* p, int n) {
    int i = blockIdx.x * 256 + threadIdx.x;
    if (i < n) p[i] = rsqrtf(p[i]);  // deg >= 1 always (self loop)
}

// out[i,:] = T[i,:] * dinv[i]^2   (self-loop term; also initializes dest buffer)
__global__ void mf_gcn_init(const float* __restrict__ T, const float* __restrict__ dinv,
                            float* __restrict__ out, int N) {
    long idx = (long)blockIdx.x * 256 + threadIdx.x;
    if (idx >= (long)N * 128) return;
    float w = dinv[(int)(idx >> 7)];
    out[idx] = T[idx] * w * w;
}

// out[dst,:] += T[src,:] * dinv[src]*dinv[dst]; 32 lanes/edge, float4/lane
__global__ void mf_edge_scatter(const float* __restrict__ T, const float* __restrict__ dinv,
                                const int* __restrict__ src, const int* __restrict__ dst,
                                float* out, int E) {
    int e = blockIdx.x * 8 + (threadIdx.x >> 5);
    if (e >= E) return;
    int lane = threadIdx.x & 31;
    int s = src[e], d = dst[e];
    float norm = dinv[s] * dinv[d];
    const float4 m = *(const float4*)(T + (size_t)s * 128 + lane * 4);
    float* o = out + (size_t)d * 128 + lane * 4;
    atomicAddF(o + 0, m.x * norm);
    atomicAddF(o + 1, m.y * norm);
    atomicAddF(o + 2, m.z * norm);
    atomicAddF(o + 3, m.w * norm);
}

__global__ void mf_bias_act(float* x, const float* __restrict__ b, long total, int relu) {
    long idx = (long)blockIdx.x * 256 + threadIdx.x;
    if (idx >= total) return;
    float v = x[idx] + b[idx & 127];
    x[idx] = relu ? fmaxf(v, 0.0f) : v;
}

__global__ void mf_pool_sum(const float* __restrict__ h, const int* __restrict__ batch,
                            float* gsum, float* cnt, int N) {
    int i = blockIdx.x * 8 + (threadIdx.x >> 5);
    if (i >= N) return;
    int lane = threadIdx.x & 31;
    int b = batch[i];
    const float4 v = *(const float4*)(h + (size_t)i * 128 + lane * 4);
    float* o = gsum + (size_t)b * 128 + lane * 4;
    atomicAddF(o + 0, v.x);
    atomicAddF(o + 1, v.y);
    atomicAddF(o + 2, v.z);
    atomicAddF(o + 3, v.w);
    if (lane == 0) atomicAddF(&cnt[b], 1.0f);
}

__global__ void mf_pool_div(float* g, const float* __restrict__ cnt, long total) {
    long idx = (long)blockIdx.x * 256 + threadIdx.x;
    if (idx >= total) return;
    g[idx] = g[idx] / fmaxf(cnt[(int)(idx >> 7)], 1.0f);
}

__global__ void mf_concat(const float* __restrict__ s, const float* __restrict__ g,
                          float* comb, int B) {
    long idx = (long)blockIdx.x * 256 + threadIdx.x;
    if (idx >= (long)B * 256) return;
    int i = (int)(idx >> 8), c = (int)(idx & 255);
    comb[idx] = (c < 128) ? s[(size_t)i * 128 + c] : g[(size_t)i * 128 + (c - 128)];
}

__global__ void mf_out(const float* __restrict__ fused, const float* __restrict__ Wo,
                       const float* __restrict__ bo, float* out, int B) {
    int idx = blockIdx.x * 256 + threadIdx.x;
    if (idx >= B * 12) return;
    int r = idx / 12, c = idx % 12;
    const float* fr = fused + (size_t)r * 128;
    float acc = bo[c];
    #pragma unroll 4
    for (int k = 0; k < 128; ++k) acc = fmaf(fr[k], Wo[k * 12 + c], acc);
    out[idx] = acc;
}

// ---------------------------------------------------------------------------
extern "C" void kernel_launch(void* const* d_in, const int* in_sizes, int n_in,
                              void* d_out, int out_size, void* d_ws, size_t ws_size,
                              hipStream_t stream) {
    (void)n_in; (void)out_size; (void)ws_size;
    const float* smiles = (const float*)d_in[0];
    const float* x      = (const float*)d_in[1];
    const int*   ei     = (const int*)d_in[2];
    const int*   batch  = (const int*)d_in[3];
    const float* Ws1 = (const float*)d_in[4];
    const float* bs1 = (const float*)d_in[5];
    const float* Ws2 = (const float*)d_in[6];
    const float* bs2 = (const float*)d_in[7];
    const float* Wg1 = (const float*)d_in[8];
    const float* bg1 = (const float*)d_in[9];
    const float* Wg2 = (const float*)d_in[10];
    const float* bg2 = (const float*)d_in[11];
    const float* Wf  = (const float*)d_in[12];
    const float* bf  = (const float*)d_in[13];
    const float* Wo  = (const float*)d_in[14];
    const float* bo  = (const float*)d_in[15];

    const int B = in_sizes[0] / 768;   // 1024 graphs
    const int N = in_sizes[1] / 128;   // 50000 nodes (multiple of 16)
    const int E = in_sizes[2] / 2;     // 800000 edges
    const int* src = ei;
    const int* dst = ei + E;

    // workspace layout (floats)
    float* ws    = (float*)d_ws;
    float* nodeA = ws;                          // N*128 : GEMM output (messages)
    float* nodeB = nodeA + (size_t)N * 128;     // N*128 : aggregated features
    float* dinv  = nodeB + (size_t)N * 128;     // N     : deg -> 1/sqrt(deg)
    float* s1    = dinv + N;                    // B*128
    float* s2    = s1 + (size_t)B * 128;        // B*128
    float* g     = s2 + (size_t)B * 128;        // B*128 (gsum, then mean in place)
    float* cnt   = g + (size_t)B * 128;         // B     (contiguous after g)
    float* comb  = cnt + B;                     // B*256
    float* fused = comb + (size_t)B * 256;      // B*128

    const dim3 blk(256);
    auto g1 = [](long n) { return dim3((unsigned)((n + 255) / 256)); };
    const dim3 gemmN((unsigned)((N / 16 + 7) / 8));
    const dim3 gemmB((unsigned)((B / 16 + 7) / 8));

    // degree / normalization
    mf_fill<<<g1(N), blk, 0, stream>>>(dinv, 1.0f, N);
    mf_edge_deg<<<g1(E), blk, 0, stream>>>(dst, dinv, E);
    mf_rsqrt<<<g1(N), blk, 0, stream>>>(dinv, N);

    // GCN layer 1
    mf_gemm_wmma<<<gemmN, blk, 0, stream>>>(x, Wg1, nodeA, N, 128);
    mf_gcn_init<<<g1((long)N * 128), blk, 0, stream>>>(nodeA, dinv, nodeB, N);
    mf_edge_scatter<<<dim3((unsigned)((E + 7) / 8)), blk, 0, stream>>>(nodeA, dinv, src, dst, nodeB, E);
    mf_bias_act<<<g1((long)N * 128), blk, 0, stream>>>(nodeB, bg1, (long)N * 128, 1);

    // GCN layer 2
    mf_gemm_wmma<<<gemmN, blk, 0, stream>>>(nodeB, Wg2, nodeA, N, 128);
    mf_gcn_init<<<g1((long)N * 128), blk, 0, stream>>>(nodeA, dinv, nodeB, N);
    mf_edge_scatter<<<dim3((unsigned)((E + 7) / 8)), blk, 0, stream>>>(nodeA, dinv, src, dst, nodeB, E);
    mf_bias_act<<<g1((long)N * 128), blk, 0, stream>>>(nodeB, bg2, (long)N * 128, 1);

    // global mean pool
    mf_fill<<<g1((long)B * 128 + B), blk, 0, stream>>>(g, 0.0f, (long)B * 128 + B);
    mf_pool_sum<<<dim3((unsigned)((N + 7) / 8)), blk, 0, stream>>>(nodeB, batch, g, cnt, N);
    mf_pool_div<<<g1((long)B * 128), blk, 0, stream>>>(g, cnt, (long)B * 128);

    // SMILES MLP
    mf_gemm_wmma<<<gemmB, blk, 0, stream>>>(smiles, Ws1, s1, B, 768);
    mf_bias_act<<<g1((long)B * 128), blk, 0, stream>>>(s1, bs1, (long)B * 128, 1);
    mf_gemm_wmma<<<gemmB, blk, 0, stream>>>(s1, Ws2, s2, B, 128);
    mf_bias_act<<<g1((long)B * 128), blk, 0, stream>>>(s2, bs2, (long)B * 128, 0);

    // fusion + output head
    mf_concat<<<g1((long)B * 256), blk, 0, stream>>>(s2, g, comb, B);
    mf_gemm_wmma<<<gemmB, blk, 0, stream>>>(comb, Wf, fused, B, 256);
    mf_bias_act<<<g1((long)B * 128), blk, 0, stream>>>(fused, bf, (long)B * 128, 1);
    mf_out<<<g1((long)B * 12), blk, 0, stream>>>(fused, Wo, bo, (float*)d_out, B);
}